// miLSTM_37374805410198
// MI455X (gfx1250) — compile-verified
//
#include <hip/hip_runtime.h>
#include <math.h>

typedef __attribute__((ext_vector_type(2))) float v2f;
typedef __attribute__((ext_vector_type(8))) float v8f;

// ---------------------------------------------------------------------------
// WMMA-based GEMV: out[r] = dot(W[r, 0:K], vec[0:K]) + bias[r]
//
// One 16-row tile per block. 128 threads = 4 wave32s; wave w accumulates the
// K-chunk [w*K/4, (w+1)*K/4) into its own 16x16 f32 accumulator via
// V_WMMA_F32_16X16X4_F32, where:
//   A[m,k] = W[rowBase+m, k]   (lane L: m = L&15, K-pair offset = 2*(L>>4))
//   B[k,n] = vec[k] for all n  (broadcast columns -> every column of D equals
//                               the GEMV result; we read column 0)
// Column 0 of C lives in lanes 0 (rows 0..7, VGPR r) and 16 (rows 8..15).
// The 4 per-wave partials are summed through LDS.
// ---------------------------------------------------------------------------
template<int K>
__global__ __launch_bounds__(128) void gemv_wmma_f32(
    const float* __restrict__ W,
    const float* __restrict__ vecBase,
    const int*   __restrict__ idx,     // optional: vec = vecBase + idx[0]*K
    const float* __restrict__ bias,
    float*       __restrict__ out,
    int nrows)
{
    __shared__ float red[64];

    const int tid  = threadIdx.x;
    const int wave = tid >> 5;
    const int lane = tid & 31;

    const int rowBase = blockIdx.x * 16;
    int row = rowBase + (lane & 15);
    if (row >= nrows) row = nrows - 1;          // clamp: keep EXEC all-1s for WMMA

    const float* vec = vecBase + (idx ? (size_t)idx[0] * (size_t)K : 0);

    const int pairOff = (lane >> 4) * 2;        // lanes 0-15: K={0,1}; 16-31: K={2,3}
    const float* wrow = W + (size_t)row * (size_t)K + pairOff;
    const float* vrow = vec + pairOff;

    const int k0 = wave * (K / 4);
    v8f c = {0.f, 0.f, 0.f, 0.f, 0.f, 0.f, 0.f, 0.f};

#pragma unroll 4
    for (int k = k0; k < k0 + K / 4; k += 4) {
        v2f a = *(const v2f*)(wrow + k);        // b64 load: W[row, k+pairOff .. +1]
        v2f b = *(const v2f*)(vrow + k);        // broadcast vec[k+pairOff .. +1]
        c = __builtin_amdgcn_wmma_f32_16x16x4_f32(
                /*neg_a=*/false, a, /*neg_b=*/false, b,
                /*c_mod=*/(short)0, c, /*reuse_a=*/false, /*reuse_b=*/false);
    }

    // Column 0 of D: lane 0 -> rows 0..7 (c[0..7]), lane 16 -> rows 8..15.
    if ((lane & 15) == 0) {
        const int half = (lane >> 4) * 8;
#pragma unroll
        for (int r = 0; r < 8; ++r)
            red[wave * 16 + half + r] = c[r];
    }
    __syncthreads();

    if (tid < 16) {
        const int r = rowBase + tid;
        if (r < nrows) {
            float s = red[tid] + red[tid + 16] + red[tid + 32] + red[tid + 48];
            out[r] = s + (bias ? bias[r] : 0.0f);
        }
    }
}

// ---------------------------------------------------------------------------
// Elementwise miLSTM gate fusion: m = a*gx*gh + b1*gx + b2*gh, then gates.
// ---------------------------------------------------------------------------
__global__ __launch_bounds__(256) void milstm_gates(
    const float* __restrict__ gx, const float* __restrict__ gh,
    const float* __restrict__ alpha, const float* __restrict__ beta1,
    const float* __restrict__ beta2, const float* __restrict__ c0,
    float* __restrict__ hx_out, float* __restrict__ cx_out)
{
    const int h = blockIdx.x * blockDim.x + threadIdx.x;
    if (h >= 1024) return;

    float m[4];
#pragma unroll
    for (int g = 0; g < 4; ++g) {
        const int j = g * 1024 + h;
        const float x = gx[j], y = gh[j];
        m[g] = alpha[j] * x * y + beta1[j] * x + beta2[j] * y;
    }
    const float fg = 1.0f / (1.0f + expf(-m[0]));
    const float ig = 1.0f / (1.0f + expf(-m[1]));
    const float og = 1.0f / (1.0f + expf(-m[2]));
    const float zt = tanhf(m[3]);

    const float cx = fg * c0[h] + ig * zt;
    const float hx = og * tanhf(cx);
    hx_out[h] = hx;
    cx_out[h] = cx;
}

// ---------------------------------------------------------------------------
// Launch: gx/gh GEMVs -> gate fusion -> decoder GEMV (205.8 MB, the hot one).
// ---------------------------------------------------------------------------
extern "C" void kernel_launch(void* const* d_in, const int* in_sizes, int n_in,
                              void* d_out, int out_size, void* d_ws, size_t ws_size,
                              hipStream_t stream) {
    constexpr int H = 1024, E = 1024, O = 50257;

    const int*   inp   = (const int*)  d_in[0];
    const float* h_0   = (const float*)d_in[1];
    const float* c_0   = (const float*)d_in[2];
    const float* emb   = (const float*)d_in[3];
    const float* Wx    = (const float*)d_in[4];
    const float* bx    = (const float*)d_in[5];
    const float* Wh    = (const float*)d_in[6];
    const float* bh    = (const float*)d_in[7];
    const float* alpha = (const float*)d_in[8];
    const float* beta1 = (const float*)d_in[9];
    const float* beta2 = (const float*)d_in[10];
    const float* Wdec  = (const float*)d_in[11];
    const float* bdec  = (const float*)d_in[12];

    float* out = (float*)d_out;          // [O]
    float* hx  = out + O;                // [H]
    float* cx  = out + O + H;            // [H]

    float* gx = (float*)d_ws;            // [4H]
    float* gh = gx + 4 * H;              // [4H]

    // gx = Wx · emb[inp] + bx   (4096 rows, 16 rows/block)
    gemv_wmma_f32<E><<<4 * H / 16, 128, 0, stream>>>(Wx, emb, inp, bx, gx, 4 * H);
    // gh = Wh · h0 + bh
    gemv_wmma_f32<H><<<4 * H / 16, 128, 0, stream>>>(Wh, h_0, nullptr, bh, gh, 4 * H);
    // gates -> hx, cx (written directly into the output buffer tail)
    milstm_gates<<<4, 256, 0, stream>>>(gx, gh, alpha, beta1, beta2, c_0, hx, cx);
    // out = Wdec · hx + bdec   (50257 rows -> 3142 tiles, 12568 waves)
    gemv_wmma_f32<H><<<(O + 15) / 16, 128, 0, stream>>>(Wdec, hx, nullptr, bdec, out, O);
}